// SAGEEncoder_55508157333630
// MI455X (gfx1250) — compile-verified
//
#include <hip/hip_runtime.h>

typedef __attribute__((ext_vector_type(2))) float v2f;
typedef __attribute__((ext_vector_type(8))) float v8f;

#define NNODES 100000
#define D 128
#define NEDGES 1600000

// ---------------------------------------------------------------- utilities
__global__ void fill_zero(float* __restrict__ p, int n) {
    int i = blockIdx.x * blockDim.x + threadIdx.x;
    int stride = gridDim.x * blockDim.x;
    for (; i < n; i += stride) p[i] = 0.0f;
}

// one wave per edge-iteration: gather 128 floats of h[src], atomic-add into agg[dst]
__global__ void scatter_kernel(const float* __restrict__ h,
                               const long long* __restrict__ src,
                               const long long* __restrict__ dst,
                               float* __restrict__ agg,
                               float* __restrict__ cnt,
                               int nE, int addCnt) {
    int lane  = threadIdx.x & 31;
    int wave  = (blockIdx.x * blockDim.x + threadIdx.x) >> 5;
    int nwave = (gridDim.x * blockDim.x) >> 5;
    for (int e = wave; e < nE; e += nwave) {
        long long s  = src[e];
        long long d0 = dst[e];
        float4 v = *((const float4*)(h + s * (long long)D) + lane);
        float* ap = agg + d0 * (long long)D + lane * 4;
        atomicAdd(ap + 0, v.x);
        atomicAdd(ap + 1, v.y);
        atomicAdd(ap + 2, v.z);
        atomicAdd(ap + 3, v.w);
        if (addCnt && lane == 0) atomicAdd(cnt + d0, 1.0f);
    }
}

__global__ void inv_kernel(const float* __restrict__ cnt,
                           float* __restrict__ inv, int n) {
    int i = blockIdx.x * blockDim.x + threadIdx.x;
    if (i < n) inv[i] = 1.0f / fmaxf(cnt[i], 1.0f);
}

// ---------------------------------------------------------------- fused GEMM
// out = ReLU( (agg .* inv) @ Wl + bl + h @ Wr )
// wave -> 16 rows x 128 cols output strip; K stepped by 4 with V_WMMA_F32_16X16X4_F32
__global__ __launch_bounds__(256) void sage_gemm(
        const float* __restrict__ agg, const float* __restrict__ inv,
        const float* __restrict__ h,
        const float* __restrict__ Wl, const float* __restrict__ bl,
        const float* __restrict__ Wr,
        float* __restrict__ out, int nRows) {
    __shared__ float sWl[D * D];
    __shared__ float sWr[D * D];
    __shared__ float sb[D];

    for (int i = threadIdx.x; i < D * D; i += blockDim.x) {
        sWl[i] = Wl[i];
        sWr[i] = Wr[i];
    }
    if (threadIdx.x < D) sb[threadIdx.x] = bl[threadIdx.x];
    __syncthreads();

    const int wave  = threadIdx.x >> 5;
    const int lane  = threadIdx.x & 31;
    const int m0    = (blockIdx.x * 8 + wave) * 16;
    if (m0 >= nRows) return;            // wave-uniform: EXEC stays all-ones for WMMA

    const int rlane = lane & 15;        // N-index / M-index within tile
    const int hi    = lane >> 4;        // half-wave selector
    const int koff  = hi * 2;           // lanes 16-31 carry K = k0+2, k0+3
    const int row_a = m0 + rlane;

    const float* arow = agg + (long long)row_a * D;
    const float* hrow = h   + (long long)row_a * D;
    const float sinv  = inv[row_a];

    v8f acc[8] = {};                    // 8 tiles of 16x16 f32 accumulators

    for (int k0 = 0; k0 < D; k0 += 4) {
        float2 am = *(const float2*)(arow + k0 + koff);
        float2 ah = *(const float2*)(hrow + k0 + koff);
        v2f a_mean; a_mean.x = am.x * sinv; a_mean.y = am.y * sinv;
        v2f a_h;    a_h.x    = ah.x;        a_h.y    = ah.y;
#pragma unroll
        for (int nt = 0; nt < 8; nt++) {
            const int n = nt * 16 + rlane;
            v2f bL, bR;
            bL.x = sWl[(k0 + koff) * D + n];
            bL.y = sWl[(k0 + koff + 1) * D + n];
            bR.x = sWr[(k0 + koff) * D + n];
            bR.y = sWr[(k0 + koff + 1) * D + n];
            acc[nt] = __builtin_amdgcn_wmma_f32_16x16x4_f32(
                false, a_mean, false, bL, (short)0, acc[nt], false, false);
            acc[nt] = __builtin_amdgcn_wmma_f32_16x16x4_f32(
                false, a_h,    false, bR, (short)0, acc[nt], false, false);
        }
    }

    // epilogue: bias + ReLU + store (two contiguous 64B row-segments per store op)
#pragma unroll
    for (int nt = 0; nt < 8; nt++) {
        const int col  = nt * 16 + rlane;
        const float bb = sb[col];
#pragma unroll
        for (int v = 0; v < 8; v++) {
            const int row = m0 + v + hi * 8;
            float val = acc[nt][v] + bb;
            out[(long long)row * D + col] = fmaxf(val, 0.0f);
        }
    }
}

// ---------------------------------------------------------------- launcher
extern "C" void kernel_launch(void* const* d_in, const int* in_sizes, int n_in,
                              void* d_out, int out_size, void* d_ws, size_t ws_size,
                              hipStream_t stream) {
    const float*     x   = (const float*)d_in[0];
    const long long* ei  = (const long long*)d_in[1];
    const float*     W1l = (const float*)d_in[2];
    const float*     b1l = (const float*)d_in[3];
    const float*     W1r = (const float*)d_in[4];
    const float*     W2l = (const float*)d_in[5];
    const float*     b2l = (const float*)d_in[6];
    const float*     W2r = (const float*)d_in[7];
    const long long* src = ei;
    const long long* dst = ei + NEDGES;
    float* out = (float*)d_out;

    float* agg  = (float*)d_ws;                    // 100000*128 floats
    float* h1   = agg  + (size_t)NNODES * D;       // 100000*128 floats
    float* cnt  = h1   + (size_t)NNODES * D;       // 100000 floats
    float* invv = cnt  + NNODES;                   // 100000 floats

    const dim3 b256(256);
    const int gemmBlocks = (NNODES / 16 + 7) / 8;  // 6250 M-tiles, 8 waves/block

    // ---- layer 1
    fill_zero<<<2048, b256, 0, stream>>>(agg, NNODES * D);
    fill_zero<<<32,   b256, 0, stream>>>(cnt, NNODES);
    scatter_kernel<<<2048, b256, 0, stream>>>(x, src, dst, agg, cnt, NEDGES, 1);
    inv_kernel<<<(NNODES + 255) / 256, b256, 0, stream>>>(cnt, invv, NNODES);
    sage_gemm<<<gemmBlocks, b256, 0, stream>>>(agg, invv, x, W1l, b1l, W1r, h1, NNODES);

    // ---- layer 2 (reuse degree counts)
    fill_zero<<<2048, b256, 0, stream>>>(agg, NNODES * D);
    scatter_kernel<<<2048, b256, 0, stream>>>(h1, src, dst, agg, cnt, NEDGES, 0);
    sage_gemm<<<gemmBlocks, b256, 0, stream>>>(agg, invv, h1, W2l, b2l, W2r, out, NNODES);
}